// JambaMoE_70574902608486
// MI455X (gfx1250) — compile-verified
//
#include <hip/hip_runtime.h>
#include <stdint.h>

// ---------------- problem constants ----------------
#define T_TOKENS 8192
#define HID      2048
#define INTERD   5632
#define NEXP     8
#define TILE_M   16
#define MAX_TILES (2 * T_TOKENS / TILE_M + NEXP)   // 1032 (fixed grid for graph capture)

typedef float v2f __attribute__((ext_vector_type(2)));
typedef float v8f __attribute__((ext_vector_type(8)));

// f32 WMMA: D(16x16) = A(16x4) * B(4x16) + C.  8-arg form per CDNA5 probe:
// (neg_a, A, neg_b, B, c_mod, C, reuse_a, reuse_b)
__device__ __forceinline__ v8f wmma_f32(v2f a, v2f b, v8f c) {
  return __builtin_amdgcn_wmma_f32_16x16x4_f32(false, a, false, b, (short)0, c, false, false);
}

// ---------------- kernel 1: zero output + counts ----------------
__global__ void zero_kernel(float4* __restrict__ out4, int* __restrict__ counts, size_t n4) {
  size_t i = (size_t)blockIdx.x * blockDim.x + threadIdx.x;
  if (i < n4) out4[i] = make_float4(0.f, 0.f, 0.f, 0.f);
  if (i < NEXP) counts[i] = 0;
}

// ---------------- kernel 2: router (wave per token) ----------------
__global__ void __launch_bounds__(256)
router_kernel(const float* __restrict__ x, const float* __restrict__ wr,
              int* __restrict__ topk_i, float* __restrict__ topk_w,
              int* __restrict__ counts) {
  const int lane = threadIdx.x & 31;
  const int wave = threadIdx.x >> 5;
  const int tok  = blockIdx.x * 8 + wave;
  const float4* xr = (const float4*)(x + (size_t)tok * HID);

  float acc[NEXP];
#pragma unroll
  for (int e = 0; e < NEXP; ++e) acc[e] = 0.f;

  for (int c = lane; c < HID / 4; c += 32) {
    float4 xv = xr[c];
#pragma unroll
    for (int e = 0; e < NEXP; ++e) {
      float4 wv = ((const float4*)(wr + (size_t)e * HID))[c];
      acc[e] += xv.x * wv.x + xv.y * wv.y + xv.z * wv.z + xv.w * wv.w;
    }
  }
#pragma unroll
  for (int e = 0; e < NEXP; ++e)
#pragma unroll
    for (int off = 16; off > 0; off >>= 1)
      acc[e] += __shfl_xor(acc[e], off, 32);

  if (lane == 0) {
    float mx = acc[0];
#pragma unroll
    for (int e = 1; e < NEXP; ++e) mx = fmaxf(mx, acc[e]);
    float p[NEXP], sum = 0.f;
#pragma unroll
    for (int e = 0; e < NEXP; ++e) { p[e] = __expf(acc[e] - mx); sum += p[e]; }
    int i0 = 0;
#pragma unroll
    for (int e = 1; e < NEXP; ++e) if (p[e] > p[i0]) i0 = e;     // lowest idx on ties
    int i1 = (i0 == 0) ? 1 : 0;
#pragma unroll
    for (int e = 0; e < NEXP; ++e) if (e != i0 && p[e] > p[i1]) i1 = e;
    float inv = 1.f / sum;
    topk_i[2 * tok]     = i0;  topk_w[2 * tok]     = p[i0] * inv;
    topk_i[2 * tok + 1] = i1;  topk_w[2 * tok + 1] = p[i1] * inv;
    atomicAdd(&counts[i0], 1);
    atomicAdd(&counts[i1], 1);
  }
}

// ---------------- kernel 3: scan counts -> offsets, build tile table ----------------
__global__ void scan_kernel(const int* __restrict__ counts, int* __restrict__ offsets,
                            int* __restrict__ cursor, int* __restrict__ tile_e,
                            int* __restrict__ tile_b, int* __restrict__ ntiles) {
  if (threadIdx.x != 0) return;
  int o = 0;
  for (int e = 0; e < NEXP; ++e) { offsets[e] = o; o += counts[e]; cursor[e] = 0; }
  offsets[NEXP] = o;
  int nt = 0;
  for (int e = 0; e < NEXP; ++e) {
    int nti = (counts[e] + TILE_M - 1) / TILE_M;
    for (int i = 0; i < nti; ++i) { tile_e[nt] = e; tile_b[nt] = offsets[e] + i * TILE_M; ++nt; }
  }
  *ntiles = nt;
}

// ---------------- kernel 4: scatter tokens into per-expert lists ----------------
__global__ void scatter_kernel(const int* __restrict__ topk_i, const float* __restrict__ topk_w,
                               const int* __restrict__ offsets, int* __restrict__ cursor,
                               int* __restrict__ perm_t, float* __restrict__ perm_w) {
  int t = blockIdx.x * blockDim.x + threadIdx.x;
  if (t >= T_TOKENS) return;
#pragma unroll
  for (int k = 0; k < 2; ++k) {
    int e = topk_i[2 * t + k];
    int pos = atomicAdd(&cursor[e], 1);
    int slot = offsets[e] + pos;
    perm_t[slot] = t;
    perm_w[slot] = topk_w[2 * t + k];
  }
}

// ---------------- kernel 5: fused grouped expert MLP ----------------
// Block = 16 routed tokens of one expert, 8 waves (256 threads).
// Loop over I in chunks of 64:
//   waves 0-3 compute gate slices, waves 4-7 compute up slices (WMMA, K=HID),
//   SiLU-combine into LDS h[16][64],
//   every wave accumulates its own 16x256 slice of h @ w_down (16 WMMA acc tiles).
// Epilogue: scale by routing weight, atomicAdd into out.
__global__ void __launch_bounds__(256, 1)
moe_kernel(const float* __restrict__ x,
           const float* __restrict__ wg, const float* __restrict__ wu,
           const float* __restrict__ wd,
           const int* __restrict__ tile_e, const int* __restrict__ tile_b,
           const int* __restrict__ ntiles, const int* __restrict__ offsets,
           const int* __restrict__ perm_t, const float* __restrict__ perm_w,
           float* __restrict__ out) {
  if ((int)blockIdx.x >= *ntiles) return;   // block-uniform

  extern __shared__ float smem[];
  float* xs   = smem;                        // [16][2048]  128 KB
  float* gbuf = xs + TILE_M * HID;           // [16][64]
  float* ubuf = gbuf + TILE_M * 64;          // [16][64]
  float* hbuf = ubuf + TILE_M * 64;          // [16][64]
  int*   ttok = (int*)(hbuf + TILE_M * 64);  // [16]
  float* twgt = (float*)(ttok + TILE_M);     // [16]

  const int tid   = threadIdx.x;
  const int wave  = tid >> 5;
  const int lane  = tid & 31;
  const int lrow  = lane & 15;
  const int lhigh = lane >> 4;

  const int e    = tile_e[blockIdx.x];
  const int base = tile_b[blockIdx.x];
  int nvalid = offsets[e + 1] - base;
  if (nvalid > TILE_M) nvalid = TILE_M;

  const float* Wg = wg + (size_t)e * HID * INTERD;
  const float* Wu = wu + (size_t)e * HID * INTERD;
  const float* Wd = wd + (size_t)e * INTERD * HID;

  if (tid < TILE_M) {
    int tk = (tid < nvalid) ? perm_t[base + tid] : -1;
    ttok[tid] = tk;
    twgt[tid] = (tid < nvalid) ? perm_w[base + tid] : 0.f;
  }
  __syncthreads();

  // Stage gathered x tile into LDS with CDNA5 async global->LDS copies (ASYNCcnt).
  for (int j = tid; j < TILE_M * HID / 4; j += 256) {
    int row = j >> 9;          // 512 float4 per row
    int c4  = j & 511;
    float* ldst = xs + row * HID + c4 * 4;
    int tk = ttok[row];
    if (tk >= 0) {
      uint32_t ldsa = (uint32_t)(uintptr_t)ldst;   // LDS aperture: low 32 bits = LDS byte addr
      uint64_t ga   = (uint64_t)(uintptr_t)(x + (size_t)tk * HID + c4 * 4);
      asm volatile("global_load_async_to_lds_b128 %0, %1, off"
                   :: "v"(ldsa), "v"(ga) : "memory");
    } else {
      ldst[0] = 0.f; ldst[1] = 0.f; ldst[2] = 0.f; ldst[3] = 0.f;
    }
  }
  asm volatile("s_wait_asynccnt 0x0" ::: "memory");
  __syncthreads();

  const v8f vzero = {0.f, 0.f, 0.f, 0.f, 0.f, 0.f, 0.f, 0.f};
  v8f acc[16];
#pragma unroll
  for (int t = 0; t < 16; ++t) acc[t] = vzero;

  const int    colbase = wave * 256;                      // down-proj output columns owned
  const float* W1      = (wave < 4) ? Wg : Wu;            // gate vs up duty
  const int    nsub    = (wave & 3) * 16;                 // 16-col slice inside the 64-chunk
  float*       stage   = ((wave < 4) ? gbuf : ubuf) + nsub;

  for (int ic = 0; ic < INTERD; ic += 64) {
    // ---- GEMM1: one 16x16 slice of gate or up, K = HID.
    // 4 independent accumulators so 4 WMMAs are in flight on the XDL pipe.
    v8f s0 = vzero, s1 = vzero, s2 = vzero, s3 = vzero;
    const float* Wp = W1 + (size_t)(ic + nsub + lrow);    // column, rows advance by INTERD
#pragma unroll 2
    for (int k0 = 0; k0 < HID; k0 += 16) {
      int ka = k0 + 2 * lhigh;
      v2f a0 = *(const v2f*)(xs + lrow * HID + ka);                 // ds_load_b64
      v2f b0 = { Wp[(size_t)ka * INTERD], Wp[(size_t)(ka + 1) * INTERD] };
      s0 = wmma_f32(a0, b0, s0);
      int kb = ka + 4;
      v2f a1 = *(const v2f*)(xs + lrow * HID + kb);
      v2f b1 = { Wp[(size_t)kb * INTERD], Wp[(size_t)(kb + 1) * INTERD] };
      s1 = wmma_f32(a1, b1, s1);
      int kc = ka + 8;
      v2f a2 = *(const v2f*)(xs + lrow * HID + kc);
      v2f b2 = { Wp[(size_t)kc * INTERD], Wp[(size_t)(kc + 1) * INTERD] };
      s2 = wmma_f32(a2, b2, s2);
      int kd = ka + 12;
      v2f a3 = *(const v2f*)(xs + lrow * HID + kd);
      v2f b3 = { Wp[(size_t)kd * INTERD], Wp[(size_t)(kd + 1) * INTERD] };
      s3 = wmma_f32(a3, b3, s3);
    }
    v8f g = (s0 + s1) + (s2 + s3);
#pragma unroll
    for (int r = 0; r < 8; ++r)                                     // C layout -> LDS
      stage[(r + 8 * lhigh) * 64 + lrow] = g[r];
    __syncthreads();

    // ---- h = silu(gate) * up   (fast sigmoid: single v_rcp_f32, ~1 ulp)
    for (int j = tid; j < TILE_M * 64; j += 256) {
      float gg = gbuf[j];
      float uu = ubuf[j];
      float sig = __builtin_amdgcn_rcpf(1.f + __expf(-gg));
      hbuf[j] = gg * sig * uu;
    }
    __syncthreads();

    // ---- GEMM2: acc += h(16x64) @ Wd[ic:ic+64, colbase:colbase+256]
    for (int k0 = 0; k0 < 64; k0 += 4) {
      int ka = k0 + 2 * lhigh;
      v2f a = *(const v2f*)(hbuf + lrow * 64 + ka);                 // ds_load_b64
      const float* w0 = Wd + (size_t)(ic + ka) * HID + colbase + lrow;
      const float* w1 = w0 + HID;
#pragma unroll
      for (int t = 0; t < 16; ++t) {                                // 16 independent tiles
        v2f b = { w0[t * 16], w1[t * 16] };
        acc[t] = wmma_f32(a, b, acc[t]);
      }
    }
    __syncthreads();   // protect g/u/h buffers for next chunk
  }

  // ---- epilogue: scale by routing weight, accumulate into out
  const int m0 = 8 * lhigh;
#pragma unroll
  for (int r = 0; r < 8; ++r) {
    int m  = m0 + r;
    int tk = ttok[m];
    if (tk < 0) continue;
    float wgt = twgt[m];
    float* orow = out + (size_t)tk * HID + colbase + lrow;
#pragma unroll
    for (int t = 0; t < 16; ++t)
      atomicAdd(orow + t * 16, acc[t][r] * wgt);
  }
}

// ---------------- host launcher ----------------
extern "C" void kernel_launch(void* const* d_in, const int* in_sizes, int n_in,
                              void* d_out, int out_size, void* d_ws, size_t ws_size,
                              hipStream_t stream) {
  (void)in_sizes; (void)n_in; (void)out_size; (void)ws_size;
  const float* x  = (const float*)d_in[0];
  const float* wr = (const float*)d_in[1];
  const float* wg = (const float*)d_in[2];
  const float* wu = (const float*)d_in[3];
  const float* wd = (const float*)d_in[4];
  float* out = (float*)d_out;

  // workspace layout (~272 KB total)
  char* w = (char*)d_ws;
  int*   counts  = (int*)(w + 0);
  int*   offsets = (int*)(w + 64);
  int*   cursor  = (int*)(w + 128);
  int*   ntiles  = (int*)(w + 192);
  int*   tile_e  = (int*)(w + 256);                 // MAX_TILES ints
  int*   tile_b  = (int*)(w + 256 + 4352);          // MAX_TILES ints
  int*   topk_i  = (int*)(w + 16384);               // T*2 ints
  float* topk_w  = (float*)(w + 16384 + 65536);     // T*2 floats
  int*   perm_t  = (int*)(w + 16384 + 131072);      // T*2 ints
  float* perm_w  = (float*)(w + 16384 + 196608);    // T*2 floats

  size_t n4 = (size_t)T_TOKENS * HID / 4;
  zero_kernel<<<(unsigned)((n4 + 255) / 256), 256, 0, stream>>>((float4*)out, counts, n4);
  router_kernel<<<T_TOKENS / 8, 256, 0, stream>>>(x, wr, topk_i, topk_w, counts);
  scan_kernel<<<1, 32, 0, stream>>>(counts, offsets, cursor, tile_e, tile_b, ntiles);
  scatter_kernel<<<(T_TOKENS + 255) / 256, 256, 0, stream>>>(topk_i, topk_w, offsets, cursor,
                                                             perm_t, perm_w);

  const int SMEM = (TILE_M * HID + 3 * TILE_M * 64) * 4 + TILE_M * 8;   // 143488 B (<320KB/WGP)
  hipFuncSetAttribute((const void*)moe_kernel,
                      hipFuncAttributeMaxDynamicSharedMemorySize, SMEM);
  moe_kernel<<<MAX_TILES, 256, SMEM, stream>>>(x, wg, wu, wd, tile_e, tile_b, ntiles,
                                               offsets, perm_t, perm_w, out);
}